// GCN_70111046140326
// MI455X (gfx1250) — compile-verified
//
#include <hip/hip_runtime.h>
#include <hip/hip_bf16.h>
#include <math.h>

// ---------------------------------------------------------------------------
// GCN (2x GCNConv, D=128) for MI455X / gfx1250, wave32 + WMMA f32 16x16x4.
// Round 3: software-pipelined K loop (prefetch next A + 8 B fragments while
// current 8 WMMAs execute) to kill the per-pair s_wait_dscnt load-use stalls.
// ---------------------------------------------------------------------------

typedef __attribute__((ext_vector_type(2))) float v2f;
typedef __attribute__((ext_vector_type(8))) float v8f;

#define DIM 128
#define ATOM_FEATS 9
#define ATOM_VOCAB 64
#define BOND_FEATS 3
#define BOND_VOCAB 16

// LDS strides (floats), padded for conflict-free fragment reads
#define SXS 132   // sX row stride: lane r hits banks 4r+k -> 16 distinct
#define SWP 288   // sWp pair-row stride: 2*128+32; +1 pair-row => +32 banks

// --------------------------- encoders --------------------------------------

__global__ __launch_bounds__(256) void k_atom_encode(const int* __restrict__ x,
    const float* __restrict__ tab, float* __restrict__ h, int n)
{
  int gid = blockIdx.x * 256 + threadIdx.x;
  if (gid >= n * DIM) return;
  int node = gid >> 7;
  int c = gid & (DIM - 1);
  const int* xf = x + node * ATOM_FEATS;
  float s = 0.f;
#pragma unroll
  for (int f = 0; f < ATOM_FEATS; ++f) {
    int idx = xf[f] + f * ATOM_VOCAB;
    s += tab[idx * DIM + c];
  }
  h[gid] = s;
}

__global__ __launch_bounds__(256) void k_deg_init(float* __restrict__ deg, int n)
{
  int i = blockIdx.x * 256 + threadIdx.x;
  if (i < n) deg[i] = 1.0f;   // self-loop weight
}

__global__ __launch_bounds__(256) void k_bond_deg(const int* __restrict__ ea,
    const float* __restrict__ btab, const int* __restrict__ col,
    float* __restrict__ ew, float* __restrict__ deg, int E)
{
  int e = blockIdx.x * 256 + threadIdx.x;
  if (e >= E) return;
  const int* a = ea + e * BOND_FEATS;
  float s = btab[a[0]] + btab[a[1] + BOND_VOCAB] + btab[a[2] + 2 * BOND_VOCAB];
  ew[e] = s;
  atomicAdd(&deg[col[e]], s);
}

__global__ __launch_bounds__(256) void k_dinv(const float* __restrict__ deg,
    float* __restrict__ dinv, int n)
{
  int i = blockIdx.x * 256 + threadIdx.x;
  if (i >= n) return;
  float d = deg[i];
  dinv[i] = (d > 0.f) ? rsqrtf(d) : 0.f;
}

// --------------------------- WMMA GEMM: Y = (relu?)X @ W -------------------
// Block: 256 threads (8 waves) -> 128-row panel of X, all 128 cols.
// sWp pair-interleaved: sWp[q*SWP + n*2 + p] = W[2q+p][n]  (q = k/2)
// => B fragment (W[k][n], W[k+1][n]) is one aligned 8B LDS load.

__global__ __launch_bounds__(256) void k_gemm(const float* __restrict__ X,
    const float* __restrict__ W, float* __restrict__ Y, int n, int relu_in)
{
  extern __shared__ float smem[];
  float* sX  = smem;                     // 128 x SXS
  float* sWp = smem + 128 * SXS;         // 64 pair-rows x SWP

  int tid = threadIdx.x;
  int wave = tid >> 5;
  int lane = tid & 31;
  int row0 = blockIdx.x * 128;

  // stage W pair-interleaved: 64 pair-rows x 32 col-groups of 4
  for (int i = tid; i < 64 * 32; i += 256) {
    int q  = i >> 5;
    int cg = (i & 31) << 2;
    float4 w0 = *(const float4*)(W + (2 * q) * 128 + cg);
    float4 w1 = *(const float4*)(W + (2 * q + 1) * 128 + cg);
    float* dst = sWp + q * SWP + cg * 2;
    *(float2*)(dst + 0) = make_float2(w0.x, w1.x);
    *(float2*)(dst + 2) = make_float2(w0.y, w1.y);
    *(float2*)(dst + 4) = make_float2(w0.z, w1.z);
    *(float2*)(dst + 6) = make_float2(w0.w, w1.w);
  }
  // stage X panel (guarded, optional input ReLU)
  for (int i = tid * 4; i < 128 * 128; i += 256 * 4) {
    int r = i >> 7, c = i & 127;
    int gr = row0 + r;
    float4 v = make_float4(0.f, 0.f, 0.f, 0.f);
    if (gr < n) v = *(const float4*)(X + gr * 128 + c);
    if (relu_in) {
      v.x = fmaxf(v.x, 0.f); v.y = fmaxf(v.y, 0.f);
      v.z = fmaxf(v.z, 0.f); v.w = fmaxf(v.w, 0.f);
    }
    *(float4*)(sX + r * SXS + c) = v;
  }
  __syncthreads();

  // fragment indices (ISA 16x4 f32 A layout / 16x16 f32 C layout)
  int m  = lane & 15;            // row (A) / col (B,C)
  int kb = (lane >> 4) << 1;     // K sub-offset: 0 or 2
  const float* ax = sX + (wave * 16 + m) * SXS + kb;       // + k0
  const float* bx = sWp + (kb >> 1) * SWP + m * 2;         // + (k0/2)*SWP + ct*32

  v8f acc[8] = {};               // 8 col-tiles x 8 VGPRs

  // --- software-pipelined K loop: prefetch next step during current WMMAs ---
  v2f a = *(const v2f*)(ax);
  v2f b[8];
#pragma unroll
  for (int ct = 0; ct < 8; ++ct) b[ct] = *(const v2f*)(bx + ct * 32);

#pragma unroll 2
  for (int k0 = 0; k0 < 128; k0 += 4) {
    int kn = (k0 + 4) & 127;                     // wrap: harmless over-prefetch
    v2f a_n = *(const v2f*)(ax + kn);
    const float* bk = bx + (kn >> 1) * SWP;
    v2f b_n[8];
#pragma unroll
    for (int ct = 0; ct < 8; ++ct) b_n[ct] = *(const v2f*)(bk + ct * 32);

#pragma unroll
    for (int ct = 0; ct < 8; ++ct) {
      acc[ct] = __builtin_amdgcn_wmma_f32_16x16x4_f32(
          /*neg_a=*/false, a, /*neg_b=*/false, b[ct],
          /*c_mod=*/(short)0, acc[ct],
          /*reuse_a=*/false, /*reuse_b=*/false);
    }
    a = a_n;
#pragma unroll
    for (int ct = 0; ct < 8; ++ct) b[ct] = b_n[ct];
  }

  // C/D layout: VGPR j, lanes 0-15 -> M=j, lanes 16-31 -> M=j+8; N = lane&15
  int rl = wave * 16 + ((lane >> 4) << 3);
  if (row0 + 128 <= n) {
    // full block: unguarded stores
#pragma unroll
    for (int ct = 0; ct < 8; ++ct) {
#pragma unroll
      for (int j = 0; j < 8; ++j) {
        Y[(row0 + rl + j) * 128 + ct * 16 + m] = acc[ct][j];
      }
    }
  } else {
#pragma unroll
    for (int ct = 0; ct < 8; ++ct) {
#pragma unroll
      for (int j = 0; j < 8; ++j) {
        int gr = row0 + rl + j;
        if (gr < n) Y[gr * 128 + ct * 16 + m] = acc[ct][j];
      }
    }
  }
}

// --------------------- aggregation: self-loop init + edge scatter ----------

__global__ __launch_bounds__(256) void k_init_out(const float* __restrict__ t,
    const float* __restrict__ dinv, const float* __restrict__ bias,
    float* __restrict__ out, int n)
{
  int gid = blockIdx.x * 256 + threadIdx.x;
  if (gid >= n * DIM) return;
  int node = gid >> 7;
  int c = gid & (DIM - 1);
  float dv = dinv[node];
  out[gid] = dv * dv * t[gid] + bias[c];   // self-loop norm = dinv^2 * 1
}

// one wave per edge, each lane owns 4 channels
__global__ __launch_bounds__(256) void k_scatter(const float* __restrict__ t,
    const int* __restrict__ row, const int* __restrict__ col,
    const float* __restrict__ ew, const float* __restrict__ dinv,
    float* __restrict__ out, int E)
{
  int gid = blockIdx.x * 256 + threadIdx.x;
  int e = gid >> 5;
  int lane = gid & 31;
  if (e >= E) return;
  int r = row[e], c = col[e];
  float nrm = dinv[r] * ew[e] * dinv[c];
  float4 v = *(const float4*)(t + r * 128 + lane * 4);
  float* dst = out + c * 128 + lane * 4;
  atomicAdd(dst + 0, nrm * v.x);
  atomicAdd(dst + 1, nrm * v.y);
  atomicAdd(dst + 2, nrm * v.z);
  atomicAdd(dst + 3, nrm * v.w);
}

// --------------------- final linear + sigmoid (wave-per-node) --------------

__global__ __launch_bounds__(256) void k_final(const float* __restrict__ h,
    const float* __restrict__ lw, const float* __restrict__ lb,
    float* __restrict__ out, int n)
{
  int gid = blockIdx.x * 256 + threadIdx.x;
  int node = gid >> 5;
  int lane = gid & 31;
  if (node >= n) return;
  float4 v = *(const float4*)(h + node * 128 + lane * 4);
  float4 w = *(const float4*)(lw + lane * 4);
  float s = v.x * w.x + v.y * w.y + v.z * w.z + v.w * w.w;
#pragma unroll
  for (int off = 16; off > 0; off >>= 1)
    s += __shfl_xor(s, off, 32);
  if (lane == 0)
    out[node] = 1.f / (1.f + __expf(-(s + lb[0])));
}

// --------------------------------- launch ----------------------------------

extern "C" void kernel_launch(void* const* d_in, const int* in_sizes, int n_in,
                              void* d_out, int out_size, void* d_ws, size_t ws_size,
                              hipStream_t stream)
{
  const int*   x          = (const int*)d_in[0];
  const int*   edge_index = (const int*)d_in[1];
  const int*   edge_attr  = (const int*)d_in[2];
  const float* atom_table = (const float*)d_in[3];
  const float* bond_table = (const float*)d_in[4];
  const float* W1   = (const float*)d_in[5];
  const float* b1   = (const float*)d_in[6];
  const float* W2   = (const float*)d_in[7];
  const float* b2   = (const float*)d_in[8];
  const float* linW = (const float*)d_in[9];
  const float* linB = (const float*)d_in[10];

  const int N = in_sizes[0] / ATOM_FEATS;
  const int E = in_sizes[2] / BOND_FEATS;
  const int* row = edge_index;
  const int* col = edge_index + E;

  // carve workspace (all buffers fully written before read each call)
  char* p = (char*)d_ws;
  auto carve = [&](size_t bytes) -> char* {
    char* q = p; p += (bytes + 255) & ~(size_t)255; return q;
  };
  float* h0   = (float*)carve((size_t)N * DIM * sizeof(float));
  float* t    = (float*)carve((size_t)N * DIM * sizeof(float));
  float* h1   = (float*)carve((size_t)N * DIM * sizeof(float));
  float* ew   = (float*)carve((size_t)E * sizeof(float));
  float* deg  = (float*)carve((size_t)N * sizeof(float));
  float* dinv = (float*)carve((size_t)N * sizeof(float));
  float* out  = (float*)d_out;

  const int ND = N * DIM;
  dim3 blk(256);
  const size_t lds = (size_t)(128 * SXS + 64 * SWP) * sizeof(float);
  const int gemm_blocks = (N + 127) / 128;

  // encoders + normalization
  k_atom_encode<<<(ND + 255) / 256, blk, 0, stream>>>(x, atom_table, h0, N);
  k_deg_init<<<(N + 255) / 256, blk, 0, stream>>>(deg, N);
  k_bond_deg<<<(E + 255) / 256, blk, 0, stream>>>(edge_attr, bond_table, col, ew, deg, E);
  k_dinv<<<(N + 255) / 256, blk, 0, stream>>>(deg, dinv, N);

  // layer 1: t = h0 @ W1 ; h1 = scatter(norm * t) + dinv^2*t + b1
  k_gemm<<<gemm_blocks, blk, lds, stream>>>(h0, W1, t, N, 0);
  k_init_out<<<(ND + 255) / 256, blk, 0, stream>>>(t, dinv, b1, h1, N);
  k_scatter<<<(E * 32 + 255) / 256, blk, 0, stream>>>(t, row, col, ew, dinv, h1, E);

  // layer 2: ReLU folded into GEMM input staging
  k_gemm<<<gemm_blocks, blk, lds, stream>>>(h1, W2, t, N, 1);
  k_init_out<<<(ND + 255) / 256, blk, 0, stream>>>(t, dinv, b2, h0, N);
  k_scatter<<<(E * 32 + 255) / 256, blk, 0, stream>>>(t, row, col, ew, dinv, h0, E);

  // final linear + sigmoid
  k_final<<<(N * 32 + 255) / 256, blk, 0, stream>>>(h0, linW, linB, out, N);
}